// SequentialROUNDModel_9861244912247
// MI455X (gfx1250) — compile-verified
//
#include <hip/hip_runtime.h>
#include <hip/hip_bf16.h>

// Problem constants (match reference)
#define B_  256
#define S_  4096
#define H_  64
#define CHUNK 256           // S positions per chunk
#define CNUM  (S_ / CHUNK)  // 16 chunks
#define TWO_PI     6.283185307179586f
#define INV_TWO_PI 0.15915494309189535f

typedef __attribute__((ext_vector_type(2))) float v2f;
typedef __attribute__((ext_vector_type(8))) float v8f;

// omega * atan2(sin(p), cos(p)) with p = x*we + be, computed as angle wrap:
// delta = p - 2*pi*rint(p/(2*pi));  contribution = om*p - (om*2pi)*rint(p/(2*pi))
__device__ __forceinline__ float wrap_contrib(float x, float we, float be,
                                              float om, float om2pi) {
    float p = __builtin_fmaf(x, we, be);
    float r = __builtin_rintf(p * INV_TWO_PI);   // v_rndne_f32
    return __builtin_fmaf(om, p, -(om2pi * r));
}

// ---------------- Pass 1: per-chunk sums of omega*delta ----------------
// grid (CNUM, B_), block 64 (one thread per h)
__global__ __launch_bounds__(64)
void k_chunksum(const float* __restrict__ x, const float* __restrict__ We,
                const float* __restrict__ be, const float* __restrict__ om,
                float* __restrict__ ws0) {
    __shared__ float lx[CHUNK];
    const int c = blockIdx.x, b = blockIdx.y, h = threadIdx.x;
    const float* xp = x + (size_t)b * S_ + (size_t)c * CHUNK;
    for (int i = h; i < CHUNK; i += 64) lx[i] = xp[i];
    __syncthreads();
    const float we = We[h], bb = be[h], o = om[h], o2p = o * TWO_PI;
    float s = 0.f;
#pragma unroll 8
    for (int i = 0; i < CHUNK; ++i) s += wrap_contrib(lx[i], we, bb, o, o2p);
    ws0[((size_t)b * CNUM + c) * H_ + h] = s;
}

// ---------------- Pass 2: exclusive scan of chunk sums -----------------
// grid B_, block 64
__global__ __launch_bounds__(64)
void k_scanchunks(const float* __restrict__ ws0, float* __restrict__ ws1) {
    const int b = blockIdx.x, h = threadIdx.x;
    float run = 0.f;
#pragma unroll
    for (int c = 0; c < CNUM; ++c) {
        const size_t idx = ((size_t)b * CNUM + c) * H_ + h;
        const float v = ws0[idx];
        ws1[idx] = run;
        run += v;
    }
}

// ---------------- Pass 3: WMMA inclusive scan, writes Hseq -------------
// grid (CNUM, B_), block 128 (4 waves; wave w handles h-tile [16w,16w+16))
// Per 16x16 tile: D = Ltri(16x16) @ Delta(16x16) + prefix, via 4 chained
// v_wmma_f32_16x16x4_f32.  Carry row D(15,:) broadcast with ds_bpermute.
__global__ __launch_bounds__(128)
void k_scan_wmma(const float* __restrict__ x, const float* __restrict__ We,
                 const float* __restrict__ be, const float* __restrict__ om,
                 const float* __restrict__ ws1, float* __restrict__ Hseq) {
    __shared__ float lx[CHUNK];
    const int c    = blockIdx.x, b = blockIdx.y;
    const int tid  = threadIdx.x;
    const int lane = tid & 31;
    const int wv   = tid >> 5;       // h-tile index 0..3
    const int n    = lane & 15;      // column (h within tile) / row M within half
    const int half = lane >> 4;      // 0: K even-pair base 0, 1: base 2
    const int h    = wv * 16 + n;
    const int kb   = half * 2;

    const float* xp = x + (size_t)b * S_ + (size_t)c * CHUNK;
    for (int i = tid; i < CHUNK; i += 128) lx[i] = xp[i];

    const float we = We[h], bb = be[h], o = om[h], o2p = o * TWO_PI;

    // A-matrix (16x4 f32) per-lane constants for the 4 K-slices of Ltri:
    // layout: v0 = (M=n, K=kb), v1 = (M=n, K=kb+1); A(i,k)=1 iff k<=i
    v2f A[4];
#pragma unroll
    for (int kk = 0; kk < 4; ++kk) {
        A[kk].x = ((kk * 4 + kb)     <= n) ? 1.f : 0.f;
        A[kk].y = ((kk * 4 + kb + 1) <= n) ? 1.f : 0.f;
    }

    // running prefix (exclusive across chunks, inclusive across segments)
    float P = ws1[((size_t)b * CNUM + c) * H_ + h];
    __syncthreads();

    // lane's output base for D register r=0 of segment 0
    float* outp = Hseq + ((size_t)b * S_ + (size_t)c * CHUNK + half * 8) * H_
                       + wv * 16 + n;

    for (int seg = 0; seg < CHUNK / 16; ++seg) {
        const int sb = seg * 16 + kb;
        v8f d = {P, P, P, P, P, P, P, P};
#pragma unroll
        for (int kk = 0; kk < 4; ++kk) {
            v2f bv;   // B (4x16): v0 = (K=kb, N=n), v1 = (K=kb+1, N=n)
            bv.x = wrap_contrib(lx[sb + kk * 4],     we, bb, o, o2p);
            bv.y = wrap_contrib(lx[sb + kk * 4 + 1], we, bb, o, o2p);
            d = __builtin_amdgcn_wmma_f32_16x16x4_f32(
                    false, A[kk], false, bv, (short)0, d, false, false);
        }
        // D reg r holds (M = r + 8*half, N = n) -> Hseq[b, s0+M, h]
#pragma unroll
        for (int r = 0; r < 8; ++r) outp[(size_t)r * H_] = d[r];
        outp += (size_t)16 * H_;

        // carry: row 15 lives in d[7] of lanes 16..31 (lane 16+n holds col n)
        int bits = __builtin_amdgcn_ds_bpermute((16 + n) << 2,
                                                __float_as_int(d[7]));
        P = __int_as_float(bits);
    }
}

// ---------------- Pass 4: out = feat @ W_r^T + b_r ---------------------
// grid B_, block 64
__global__ __launch_bounds__(64)
void k_out(const float* __restrict__ Hseq, const float* __restrict__ Wr,
           const float* __restrict__ br, float* __restrict__ out) {
    __shared__ float red[64];
    const int b = blockIdx.x, h = threadIdx.x;
    const float ph = Hseq[((size_t)b * S_ + (S_ - 1)) * H_ + h];
    red[h] = cosf(ph) * Wr[h] + sinf(ph) * Wr[H_ + h] + ph * Wr[2 * H_ + h];
    __syncthreads();
#pragma unroll
    for (int off = 32; off > 0; off >>= 1) {
        if (h < off) red[h] += red[h + off];
        __syncthreads();
    }
    if (h == 0) out[b] = red[0] + br[0];
}

extern "C" void kernel_launch(void* const* d_in, const int* in_sizes, int n_in,
                              void* d_out, int out_size, void* d_ws, size_t ws_size,
                              hipStream_t stream) {
    const float* x  = (const float*)d_in[0];
    const float* We = (const float*)d_in[1];
    const float* be = (const float*)d_in[2];
    const float* om = (const float*)d_in[3];
    const float* Wr = (const float*)d_in[4];
    const float* br = (const float*)d_in[5];

    float* out  = (float*)d_out;        // 256 floats
    float* Hseq = out + B_;             // 256*4096*64 floats

    float* ws0 = (float*)d_ws;                          // B*CNUM*H chunk sums
    float* ws1 = ws0 + (size_t)B_ * CNUM * H_;          // exclusive prefixes

    k_chunksum <<<dim3(CNUM, B_), 64,  0, stream>>>(x, We, be, om, ws0);
    k_scanchunks<<<B_,            64,  0, stream>>>(ws0, ws1);
    k_scan_wmma<<<dim3(CNUM, B_), 128, 0, stream>>>(x, We, be, om, ws1, Hseq);
    k_out      <<<B_,             64,  0, stream>>>(Hseq, Wr, br, out);
}